// SAGE_19353122635776
// MI455X (gfx1250) — compile-verified
//
#include <hip/hip_runtime.h>

typedef float v2f __attribute__((ext_vector_type(2)));
typedef float v8f __attribute__((ext_vector_type(8)));
typedef unsigned int u32x4 __attribute__((ext_vector_type(4)));
typedef int i32x4 __attribute__((ext_vector_type(4)));
typedef int i32x8 __attribute__((ext_vector_type(8)));

#define DFEAT 64
#define HSTRIDE 68   // padded LDS row stride (float units) to spread banks

#if defined(__HIP_DEVICE_COMPILE__) && __has_builtin(__builtin_amdgcn_tensor_load_to_lds)
#  if __has_include(<hip/amd_detail/amd_gfx1250_TDM.h>)
#    define SAGE_TDM_6ARG 1
#  else
#    define SAGE_TDM_5ARG 1
#  endif
#endif

// ---------------- degree / normalizer ----------------
__global__ void deg_kernel(const int* __restrict__ dst, int* __restrict__ deg, int nE) {
    int e = blockIdx.x * blockDim.x + threadIdx.x;
    if (e < nE) atomicAdd(&deg[dst[e]], 1);
}

__global__ void inv_kernel(const int* __restrict__ deg, float* __restrict__ inv, int nN) {
    int n = blockIdx.x * blockDim.x + threadIdx.x;
    if (n < nN) inv[n] = 1.0f / (float)(deg[n] + 1);
}

// ---------------- edge scatter-add: agg[dst] += h[src] ----------------
// one thread per (edge, 16-feature chunk): 4 x float4 gather + 16 f32 atomics
__global__ void aggregate_kernel(const float4* __restrict__ h4,
                                 const int* __restrict__ src,
                                 const int* __restrict__ dst,
                                 float* __restrict__ agg, int nE) {
    int t = blockIdx.x * blockDim.x + threadIdx.x;
    int e = t >> 2;
    int c = t & 3;
    if (e >= nE) return;
    int s = src[e];
    int d = dst[e];
    const float4* hp = h4 + (size_t)s * (DFEAT / 4) + c * 4;
    float* ap = agg + (size_t)d * DFEAT + c * 16;
#pragma unroll
    for (int j = 0; j < 4; ++j) {
        float4 v = hp[j];
        unsafeAtomicAdd(ap + j * 4 + 0, v.x);
        unsafeAtomicAdd(ap + j * 4 + 1, v.y);
        unsafeAtomicAdd(ap + j * 4 + 2, v.z);
        unsafeAtomicAdd(ap + j * 4 + 3, v.w);
    }
}

// ---------------- fused normalize + GEMM (+bias, +optional relu) ----------------
// block = 256 threads = 8 waves; each wave computes a 16(node) x 64(out) tile
// via V_WMMA_F32_16X16X4_F32, K-loop of 16, 4 N-tiles. W is staged into LDS by
// a single Tensor-Data-Mover descriptor issued from wave 0 (scalar branch:
// TDM ignores EXEC, so EXEC-masked waves would still issue it).
template <bool RELU>
__global__ __launch_bounds__(256)
void combine_gemm_kernel(const float* __restrict__ h,
                         const float* __restrict__ agg,
                         const float* __restrict__ inv,
                         const float* __restrict__ W,
                         const float* __restrict__ bias,
                         float* __restrict__ out, int nN) {
    __shared__ float Wl[DFEAT * DFEAT];         // 16 KB
    __shared__ float bl[DFEAT];
    __shared__ float hN[8][16 * HSTRIDE];       // 34 KB, per-wave 16x64 padded tiles

    const int t    = threadIdx.x;
    const int wave = t >> 5;
    const int lane = t & 31;
    const int node0 = blockIdx.x * 128 + wave * 16;

#if defined(SAGE_TDM_5ARG) || defined(SAGE_TDM_6ARG)
    // ---- TDM stage of W: one descriptor, issued once per workgroup ----
    if (__builtin_amdgcn_readfirstlane(wave) == 0) {
        const unsigned long long ga = (unsigned long long)(const void*)W;
        const unsigned ldsb = (unsigned)(unsigned long long)(void*)&Wl[0];
        // D# group0: count=1 | lds_addr | global_addr[56:0] | type=2
        u32x4 g0;
        g0.x = 1u;                                   // count=1, user mode
        g0.y = ldsb;                                 // lds_addr (bytes)
        g0.z = (unsigned)(ga & 0xFFFFFFFFull);       // global_addr[31:0]
        g0.w = (unsigned)((ga >> 32) & 0x1FFFFFFull) // global_addr[56:32]
               | (2u << 30);                         // type=2 ("image")
        // D# group1: data_size=4B, tensor 64x64, tile 64x64, dim0_stride=64
        i32x8 g1;
        g1[0] = (int)(2u << 16);                     // workgroup_mask=0, data_size=2 (4B)
        g1[1] = (int)(64u << 16);                    // tensor_dim0[15:0]=64 (bits 63:48)
        g1[2] = (int)(64u << 16);                    // tensor_dim1[15:0]=64 (bits 95:80)
        g1[3] = (int)(64u << 16);                    // tile_dim0=64 (bits 127:112)
        g1[4] = 64;                                  // tile_dim1=64, tile_dim2=0
        g1[5] = 64;                                  // tensor_dim0_stride[31:0]=64
        g1[6] = (int)(4096u << 16);                  // tensor_dim1_stride[15:0]=4096
        g1[7] = 0;                                   // tensor_dim1_stride[47:16]=0
        i32x4 gz = {0, 0, 0, 0};
#  if defined(SAGE_TDM_6ARG)
        i32x8 gz8 = {0, 0, 0, 0, 0, 0, 0, 0};
        __builtin_amdgcn_tensor_load_to_lds(g0, g1, gz, gz, gz8, 0);
#  else
        __builtin_amdgcn_tensor_load_to_lds(g0, g1, gz, gz, 0);
#  endif
        __builtin_amdgcn_s_wait_tensorcnt(0);
    }
    if (t < 16) ((float4*)bl)[t] = ((const float4*)bias)[t];
#else
    // ---- fallback: float4 stage of W (4096 floats) and bias ----
    {
        const float4* W4 = (const float4*)W;
        float4* Wl4 = (float4*)Wl;
#pragma unroll
        for (int i = 0; i < 4; ++i) Wl4[t + i * 256] = W4[t + i * 256];
        if (t < 16) ((float4*)bl)[t] = ((const float4*)bias)[t];
    }
#endif

    // stage normalized tile: hN = (agg + h) * inv  (clamped node index; no divergence)
    {
        const int row = lane & 15;
        const int cb  = (lane >> 4) * 32;
        int gn = node0 + row;
        if (gn >= nN) gn = nN - 1;
        const float iv = inv[gn];
        const float4* a4 = (const float4*)(agg + (size_t)gn * DFEAT + cb);
        const float4* x4 = (const float4*)(h   + (size_t)gn * DFEAT + cb);
        float* dp = &hN[wave][row * HSTRIDE + cb];
#pragma unroll
        for (int j = 0; j < 8; ++j) {
            float4 a = a4[j];
            float4 x = x4[j];
            float4 r;
            r.x = (a.x + x.x) * iv;
            r.y = (a.y + x.y) * iv;
            r.z = (a.z + x.z) * iv;
            r.w = (a.w + x.w) * iv;
            *(float4*)(dp + j * 4) = r;
        }
    }
    __syncthreads();

    // WMMA: D[16x64] = A[16x64] * W[64x64]
    v8f acc[4] = {};
    const int row   = lane & 15;            // A: M index, B: N index within tile
    const int khalf = (lane >> 4) * 2;      // lanes 16-31 carry K+2,K+3
    const float* hrow = &hN[wave][row * HSTRIDE];
#pragma unroll
    for (int k0 = 0; k0 < DFEAT; k0 += 4) {
        const int kb = k0 + khalf;
        v2f A;
        A.x = hrow[kb];
        A.y = hrow[kb + 1];
#pragma unroll
        for (int j = 0; j < 4; ++j) {
            v2f B;
            B.x = Wl[kb * DFEAT + j * 16 + row];
            B.y = Wl[(kb + 1) * DFEAT + j * 16 + row];
            acc[j] = __builtin_amdgcn_wmma_f32_16x16x4_f32(
                false, A, false, B, (short)0, acc[j], false, false);
        }
    }

    // epilogue: acc[j][r] = D[(lane/16)*8 + r][j*16 + lane%16]
    const int mb = (lane >> 4) * 8;
    const int n_ = lane & 15;
#pragma unroll
    for (int j = 0; j < 4; ++j) {
        const float bj = bl[j * 16 + n_];
#pragma unroll
        for (int r = 0; r < 8; ++r) {
            const int gn = node0 + mb + r;
            if (gn < nN) {
                float v = acc[j][r] + bj;
                if (RELU) v = fmaxf(v, 0.0f);
                out[(size_t)gn * DFEAT + j * 16 + n_] = v;
            }
        }
    }
}

// ---------------- driver ----------------
extern "C" void kernel_launch(void* const* d_in, const int* in_sizes, int n_in,
                              void* d_out, int out_size, void* d_ws, size_t ws_size,
                              hipStream_t stream) {
    const float* x   = (const float*)d_in[0];
    const int*   src = (const int*)d_in[1];
    const int*   dst = (const int*)d_in[2];
    const float* W0  = (const float*)d_in[3];
    const float* b0  = (const float*)d_in[4];
    const float* W1  = (const float*)d_in[5];
    const float* b1  = (const float*)d_in[6];
    const float* W2  = (const float*)d_in[7];
    const float* b2  = (const float*)d_in[8];
    float* out = (float*)d_out;

    const int nN = in_sizes[0] / DFEAT;
    const int nE = in_sizes[1];

    char* ws = (char*)d_ws;
    size_t off = 0;
    auto carve = [&](size_t bytes) -> void* {
        void* p = ws + off;
        off = (off + bytes + 255) & ~(size_t)255;
        return p;
    };
    int*   deg = (int*)carve((size_t)nN * 4);
    float* inv = (float*)carve((size_t)nN * 4);
    float* agg = (float*)carve((size_t)nN * DFEAT * 4);
    float* h1  = (float*)carve((size_t)nN * DFEAT * 4);
    float* h2  = (float*)carve((size_t)nN * DFEAT * 4);

    const int threads = 256;
    const dim3 degGrid((nE + threads - 1) / threads);
    const dim3 invGrid((nN + threads - 1) / threads);
    const dim3 aggGrid((nE * 4 + threads - 1) / threads);
    const dim3 gemGrid((nN + 127) / 128);
    const size_t aggBytes = (size_t)nN * DFEAT * 4;

    // degree + normalizer (constant across layers)
    hipMemsetAsync(deg, 0, (size_t)nN * 4, stream);
    deg_kernel<<<degGrid, threads, 0, stream>>>(dst, deg, nE);
    inv_kernel<<<invGrid, threads, 0, stream>>>(deg, inv, nN);

    // layer 0: x -> h1 (relu)
    hipMemsetAsync(agg, 0, aggBytes, stream);
    aggregate_kernel<<<aggGrid, threads, 0, stream>>>((const float4*)x, src, dst, agg, nE);
    combine_gemm_kernel<true><<<gemGrid, threads, 0, stream>>>(x, agg, inv, W0, b0, h1, nN);

    // layer 1: h1 -> h2 (relu)
    hipMemsetAsync(agg, 0, aggBytes, stream);
    aggregate_kernel<<<aggGrid, threads, 0, stream>>>((const float4*)h1, src, dst, agg, nE);
    combine_gemm_kernel<true><<<gemGrid, threads, 0, stream>>>(h1, agg, inv, W1, b1, h2, nN);

    // layer 2: h2 -> out (no relu)
    hipMemsetAsync(agg, 0, aggBytes, stream);
    aggregate_kernel<<<aggGrid, threads, 0, stream>>>((const float4*)h2, src, dst, agg, nE);
    combine_gemm_kernel<false><<<gemGrid, threads, 0, stream>>>(h2, agg, inv, W2, b2, out, nN);
}